// MPNNSummarizer_31456340476251
// MI455X (gfx1250) — compile-verified
//
#include <hip/hip_runtime.h>
#include <hip/hip_bf16.h>
#include <math.h>

#define N_NODES 100000
#define N_EDGES 600000
#define N_GRAPHS 64
#define D 128

typedef __attribute__((ext_vector_type(2))) float v2f;
typedef __attribute__((ext_vector_type(8))) float v8f;

// ---------------------------------------------------------------------------
// GEMM: out[n][j] = sum_k act(H[n][k]) * W[j][k] + bias[j]
// act = relu if relu_input, else identity.
// Uses V_WMMA_F32_16X16X4_F32 (CDNA5 fp32 WMMA), K=128 -> 32 chained WMMAs.
// Block: 256 threads = 8 wave32s. Each block computes a 16-row x 128-col tile;
// wave w handles columns [16w, 16w+16). A-tile staged in LDS (pitch 132 to
// avoid bank conflicts). N_NODES % 16 == 0 so EXEC is all-ones everywhere.
// ---------------------------------------------------------------------------
__global__ __launch_bounds__(256) void gemm_wmma_f32(
    const float* __restrict__ H, const float* __restrict__ W,
    const float* __restrict__ bias, float* __restrict__ out, int relu_input)
{
    __shared__ float sA[16 * 132];                 // 16 rows, padded pitch 132
    const int tid  = threadIdx.x;
    const int row0 = blockIdx.x * 16;

    // Stage A tile (16x128) into LDS, fusing input activation.
    for (int i = tid; i < 16 * 128; i += 256) {
        const int r = i >> 7, cidx = i & 127;
        float v = H[(size_t)(row0 + r) * D + cidx];
        if (relu_input) v = fmaxf(v, 0.0f);
        sA[r * 132 + cidx] = v;
    }
    __syncthreads();

    const int wave = tid >> 5;          // 0..7 -> column tile
    const int lane = tid & 31;
    const int half = lane >> 4;         // 0: K pair {0,1}; 1: K pair {2,3}
    const int mA   = lane & 15;         // A row within tile
    const int n    = (wave << 4) + (lane & 15);   // global output column

    v8f c = {};                          // fp32 accumulator (8 VGPRs)
    const float* wrow = W + (size_t)n * D;  // B[k][n] == W[n][k], contiguous in k

    #pragma unroll
    for (int k0 = 0; k0 < D; k0 += 4) {
        const int ka = k0 + 2 * half;
        v2f a, b;
        a.x = sA[mA * 132 + ka];
        a.y = sA[mA * 132 + ka + 1];
        b.x = wrow[ka];
        b.y = wrow[ka + 1];
        c = __builtin_amdgcn_wmma_f32_16x16x4_f32(
                /*neg_a=*/false, a, /*neg_b=*/false, b,
                /*c_mod=*/(short)0, c, /*reuse_a=*/false, /*reuse_b=*/false);
    }

    const float bj = bias[n];
    #pragma unroll
    for (int r = 0; r < 8; ++r) {
        const int m = r + 8 * half;      // C/D layout: vgpr r, lanes>=16 -> m+8
        out[(size_t)(row0 + m) * D + n] = c[r] + bj;
    }
}

// agg = lin  (self-loop initialization), vectorized float4 copy
__global__ __launch_bounds__(256) void copy_f4(const float4* __restrict__ src,
                                               float4* __restrict__ dst, int n4)
{
    for (int i = blockIdx.x * blockDim.x + threadIdx.x; i < n4;
         i += gridDim.x * blockDim.x)
        dst[i] = src[i];
}

// agg[col[e]][:] += lin[row[e]][:]   (one 128-thread block per edge)
__global__ __launch_bounds__(128) void scatter_add_edges(
    const float* __restrict__ lin, const int* __restrict__ row,
    const int* __restrict__ col, float* __restrict__ agg)
{
    const int e = blockIdx.x;
    const int d = threadIdx.x;
    const int r = row[e];
    const int cn = col[e];
    atomicAdd(&agg[(size_t)cn * D + d], lin[(size_t)r * D + d]);
}

__global__ __launch_bounds__(256) void zero_f32(float* __restrict__ p, int n)
{
    const int i = blockIdx.x * blockDim.x + threadIdx.x;
    if (i < n) p[i] = 0.0f;
}

// Segment-sum over sorted batch ids: each block scans 250 nodes, accumulating
// in registers, flushing to global atomics only at graph-ID boundaries.
__global__ __launch_bounds__(128) void pool_sum(
    const float* __restrict__ h, const int* __restrict__ batch,
    float* __restrict__ sums, float* __restrict__ cnts, int n_nodes)
{
    const int d    = threadIdx.x;
    const int base = blockIdx.x * 250;
    int end = base + 250;
    if (end > n_nodes) end = n_nodes;
    if (base >= end) return;

    int   curg = batch[base];
    float acc  = 0.0f;
    float cacc = 0.0f;
    for (int i = base; i < end; ++i) {
        const int g = batch[i];
        if (g != curg) {
            atomicAdd(&sums[curg * D + d], acc);
            if (d == 0) atomicAdd(&cnts[curg], cacc);
            acc = 0.0f; cacc = 0.0f; curg = g;
        }
        acc  += h[(size_t)i * D + d];
        cacc += 1.0f;
    }
    atomicAdd(&sums[curg * D + d], acc);
    if (d == 0) atomicAdd(&cnts[curg], cacc);
}

// out[g] = sigmoid( dot(sums[g]/max(cnt,1), Wc) + bc )
__global__ __launch_bounds__(64) void classifier(
    const float* __restrict__ sums, const float* __restrict__ cnts,
    const float* __restrict__ Wc, const float* __restrict__ bc,
    float* __restrict__ out)
{
    const int g = threadIdx.x;          // 64 graphs, one block
    const float inv = 1.0f / fmaxf(cnts[g], 1.0f);
    float acc = 0.0f;
    #pragma unroll 4
    for (int k = 0; k < D; ++k)
        acc += sums[g * D + k] * Wc[k];
    acc = acc * inv + bc[0];
    out[g] = 1.0f / (1.0f + expf(-acc));
}

extern "C" void kernel_launch(void* const* d_in, const int* in_sizes, int n_in,
                              void* d_out, int out_size, void* d_ws, size_t ws_size,
                              hipStream_t stream)
{
    (void)in_sizes; (void)n_in; (void)out_size; (void)ws_size;

    const float* x     = (const float*)d_in[0];
    const int*   eidx  = (const int*)d_in[1];          // [2, 600000] flat
    const int*   row   = eidx;                          // sources
    const int*   col   = eidx + N_EDGES;                // targets
    const int*   batch = (const int*)d_in[2];
    const float* Wl[4] = { (const float*)d_in[3], (const float*)d_in[5],
                           (const float*)d_in[7], (const float*)d_in[9] };
    const float* bl[4] = { (const float*)d_in[4], (const float*)d_in[6],
                           (const float*)d_in[8], (const float*)d_in[10] };
    const float* Wc = (const float*)d_in[11];
    const float* bc = (const float*)d_in[12];
    float* out = (float*)d_out;

    // Workspace carve-up (all 256B-aligned)
    const size_t nodeBytes = (size_t)N_NODES * D * sizeof(float); // 51.2 MB
    char* ws = (char*)d_ws;
    float* bufA = (float*)ws;                         // lin buffer
    float* bufB = (float*)(ws + nodeBytes);           // agg / h buffer
    float* sums = (float*)(ws + 2 * nodeBytes);       // 64*128
    float* cnts = (float*)(ws + 2 * nodeBytes + N_GRAPHS * D * sizeof(float));

    const int n4        = N_NODES * D / 4;            // float4 count
    const int copyBlks  = 2048;
    const int gemmBlks  = N_NODES / 16;               // 6250

    const float* h = x;
    for (int layer = 0; layer < 4; ++layer) {
        const int relu_in = (layer > 0) ? 1 : 0;
        // lin = act(h) @ W^T + b
        gemm_wmma_f32<<<gemmBlks, 256, 0, stream>>>(h, Wl[layer], bl[layer],
                                                    bufA, relu_in);
        // agg = lin (self loops); safe: gemm has fully consumed bufB already
        copy_f4<<<copyBlks, 256, 0, stream>>>((const float4*)bufA,
                                              (float4*)bufB, n4);
        // agg[col] += lin[row]
        scatter_add_edges<<<N_EDGES, 128, 0, stream>>>(bufA, row, col, bufB);
        h = bufB;
    }

    // Global mean pool + linear + sigmoid
    const int poolElems = N_GRAPHS * D + N_GRAPHS;
    zero_f32<<<(poolElems + 255) / 256, 256, 0, stream>>>(sums, poolElems);
    pool_sum<<<(N_NODES + 249) / 250, 128, 0, stream>>>(bufB, batch, sums, cnts,
                                                        N_NODES);
    classifier<<<1, 64, 0, stream>>>(sums, cnts, Wc, bc, out);
}